// MySupConLoss_84928683311678
// MI455X (gfx1250) — compile-verified
//
#include <hip/hip_runtime.h>

typedef __attribute__((ext_vector_type(2))) float v2f;
typedef __attribute__((ext_vector_type(8))) float v8f;

#define DDIM    128
#define KSTEPS  (DDIM / 4)
#define NWAVES  8
#define TEMP_C  0.07f
#define LOSS_SCALE 1.0f   // TEMP / BASE_TEMP

__device__ __forceinline__ const float* cf_row(const float* fi, const float* fj, int B, int r) {
    return (r < B) ? (fi + (size_t)r * DDIM) : (fj + (size_t)(r - B) * DDIM);
}
__device__ __forceinline__ int cf_lab(const int* li, const int* lj, int B, int r) {
    return (r < B) ? li[r] : lj[r - B];
}

__global__ __launch_bounds__(256)
void supcon_main(const float* __restrict__ fi, const float* __restrict__ fj,
                 const int* __restrict__ li, const int* __restrict__ lj,
                 int B, int N, int numTiles, float* __restrict__ ws)
{
    const int tid  = threadIdx.x;
    const int wave = tid >> 5;
    const int lane = tid & 31;
    const int l16  = lane & 15;
    const int half = lane >> 4;      // 0 or 1
    const int koff = half * 2;       // f32 16x4 A/B layout: lanes 16-31 hold K+2

    const int row_base = blockIdx.x * 16;

    // ---- Preload A fragments for the whole D=128 (16x4 f32 layout) ----
    const float* arow = cf_row(fi, fj, B, row_base + l16);
    v2f afrag[KSTEPS];
#pragma unroll
    for (int k = 0; k < KSTEPS; ++k)
        afrag[k] = *(const v2f*)(arow + k * 4 + koff);

    // Rows owned by this lane (C/D layout: VGPR v -> M = v + half*8)
    int rrow[8], rlab[8];
#pragma unroll
    for (int v = 0; v < 8; ++v) {
        rrow[v] = row_base + v + half * 8;
        rlab[v] = cf_lab(li, lj, B, rrow[v]);
    }

    float mr[8], sr[8], pr[8], cr[8];
#pragma unroll
    for (int v = 0; v < 8; ++v) { mr[v] = -__builtin_inff(); sr[v] = 0.f; pr[v] = 0.f; cr[v] = 0.f; }

    const float invT  = 1.0f / TEMP_C;
    const int colsPerWave = N / NWAVES;
    const int tiles  = colsPerWave / 16;
    const int cbase0 = wave * colsPerWave;

    for (int t = 0; t < tiles; ++t) {
        const int col_base = cbase0 + t * 16;
        const int c        = col_base + l16;
        const float* brow  = cf_row(fi, fj, B, c);
        const int    clab  = cf_lab(li, lj, B, c);

        v8f acc = {0.f,0.f,0.f,0.f,0.f,0.f,0.f,0.f};
#pragma unroll
        for (int k = 0; k < KSTEPS; ++k) {
            v2f b = *(const v2f*)(brow + k * 4 + koff);
            acc = __builtin_amdgcn_wmma_f32_16x16x4_f32(
                false, afrag[k], false, b, (short)0, acc, false, false);
        }

        // Online masked LSE + positive accumulation.
        // acc[v] = S[rrow[v]][c]; diag counts for max only.
#pragma unroll
        for (int v = 0; v < 8; ++v) {
            float x  = acc[v] * invT;
            float nm = fmaxf(mr[v], x);
            float e  = (c == rrow[v]) ? 0.f : __expf(x - nm);
            sr[v] = sr[v] * __expf(mr[v] - nm) + e;
            mr[v] = nm;
            bool pos = (clab == rlab[v]) && (c != rrow[v]);
            pr[v] += pos ? x  : 0.f;
            cr[v] += pos ? 1.f : 0.f;
        }
    }

    // ---- Butterfly LSE-merge across the 16 lanes of each half-wave ----
#pragma unroll
    for (int v = 0; v < 8; ++v) {
        float m = mr[v], s = sr[v], p = pr[v], cc = cr[v];
#pragma unroll
        for (int off = 1; off < 16; off <<= 1) {
            float mo = __shfl_xor(m,  off, 32);
            float so = __shfl_xor(s,  off, 32);
            float po = __shfl_xor(p,  off, 32);
            float co = __shfl_xor(cc, off, 32);
            float nm = fmaxf(m, mo);
            s = s * __expf(m - nm) + so * __expf(mo - nm);
            m = nm; p += po; cc += co;
        }
        mr[v] = m; sr[v] = s; pr[v] = p; cr[v] = cc;
    }

    __shared__ float Lm[NWAVES][16], Ls[NWAVES][16], Lp[NWAVES][16], Lc[NWAVES][16];
    if (l16 == 0) {
#pragma unroll
        for (int v = 0; v < 8; ++v) {
            int ri = v + half * 8;
            Lm[wave][ri] = mr[v]; Ls[wave][ri] = sr[v];
            Lp[wave][ri] = pr[v]; Lc[wave][ri] = cr[v];
        }
    }
    __syncthreads();

    __shared__ float rowloss[16], rowvalid[16];
    if (tid < 16) {
        float M = -__builtin_inff(), S = 0.f, P = 0.f, C = 0.f;
#pragma unroll
        for (int w = 0; w < NWAVES; ++w) {
            float mo = Lm[w][tid], so = Ls[w][tid];
            float nm = fmaxf(M, mo);
            S = S * __expf(M - nm) + so * __expf(mo - nm);
            M = nm; P += Lp[w][tid]; C += Lc[w][tid];
        }
        float lse = M + __logf(S);
        float rl = 0.f, vv = 0.f;
        if (C > 0.f) { rl = -LOSS_SCALE * (P / C - lse); vv = 1.f; }
        rowloss[tid]  = rl;
        rowvalid[tid] = vv;
    }
    __syncthreads();
    if (tid == 0) {
        float a = 0.f, b = 0.f;
#pragma unroll
        for (int i = 0; i < 16; ++i) { a += rowloss[i]; b += rowvalid[i]; }
        ws[blockIdx.x]            = a;
        ws[numTiles + blockIdx.x] = b;
    }
}

__global__ void supcon_finalize(const float* __restrict__ ws, int numTiles,
                                float* __restrict__ out)
{
    int lane = threadIdx.x;
    float a = 0.f, b = 0.f;
    for (int i = lane; i < numTiles; i += 32) { a += ws[i]; b += ws[numTiles + i]; }
#pragma unroll
    for (int off = 16; off; off >>= 1) {
        a += __shfl_xor(a, off, 32);
        b += __shfl_xor(b, off, 32);
    }
    if (lane == 0) out[0] = a / fmaxf(b, 1.f);
}

extern "C" void kernel_launch(void* const* d_in, const int* in_sizes, int n_in,
                              void* d_out, int out_size, void* d_ws, size_t ws_size,
                              hipStream_t stream)
{
    const float* fi = (const float*)d_in[0];
    const float* fj = (const float*)d_in[1];
    const int*   li = (const int*)d_in[2];
    const int*   lj = (const int*)d_in[3];
    float* out = (float*)d_out;
    float* ws  = (float*)d_ws;

    const int B        = in_sizes[2];   // 4096
    const int N        = 2 * B;         // 8192
    const int numTiles = N / 16;        // 512

    supcon_main<<<numTiles, 256, 0, stream>>>(fi, fj, li, lj, B, N, numTiles, ws);
    supcon_finalize<<<1, 32, 0, stream>>>(ws, numTiles, out);
}